// BatchedCholeskyOrthogonalization_42588895707563
// MI455X (gfx1250) — compile-verified
//
#include <hip/hip_runtime.h>

typedef float v2f __attribute__((ext_vector_type(2)));
typedef float v8f __attribute__((ext_vector_type(8)));
typedef unsigned int v4u __attribute__((ext_vector_type(4)));
typedef int v4i __attribute__((ext_vector_type(4)));
typedef int v8i __attribute__((ext_vector_type(8)));

#define NBATCH 256
#define MDIM   256   // rows per batch
#define KDIM   512   // inner dim per batch
#define EPS_C  1e-5f

#if defined(__has_builtin)
#if __has_builtin(__builtin_amdgcn_tensor_load_to_lds) && __has_builtin(__builtin_amdgcn_s_wait_tensorcnt)
#define HAVE_TDM 1
#endif
#endif

#ifdef HAVE_TDM
// ---------------------------------------------------------------------------
// Issue one TDM 2D tile load Global -> LDS (D# groups per CDNA5 ISA 8.3/8.4).
// data_size = 4 bytes; dims/strides in elements. Whole-wave (EXEC ignored).
// 6-arg builtin form: (g0:u32x4, g1:i32x8, g2:i32x4, g3:i32x4, i32x8, cpol).
// ---------------------------------------------------------------------------
__device__ __forceinline__ void tdm_load_2d_f32(
    unsigned lds_byte_off, const void* gaddr,
    unsigned tile_d0, unsigned tile_d1,
    unsigned tensor_d0, unsigned tensor_d1,
    unsigned long long stride0_elems)
{
    unsigned long long ga = (unsigned long long)(size_t)gaddr;
    v4u g0;
    g0.x = 1u;                                     // count=1, user mode, no gather
    g0.y = lds_byte_off;                           // lds_addr [63:32]
    g0.z = (unsigned)(ga & 0xFFFFFFFFu);           // global_addr [95:64]
    g0.w = (unsigned)((ga >> 32) & 0x01FFFFFFu)    // global_addr [120:96]
         | (2u << 30);                             // type=2 ("image") [127:126]
    v8i g1;
    g1[0] = (int)(2u << 16);                       // wg_mask=0, data_size=2 (4B)
    g1[1] = (int)((tensor_d0 & 0xFFFFu) << 16);    // tensor_dim0 low16 -> bits[63:48]
    g1[2] = (int)((tensor_d0 >> 16) | ((tensor_d1 & 0xFFFFu) << 16));
    g1[3] = (int)((tensor_d1 >> 16) | ((tile_d0 & 0xFFFFu) << 16));
    g1[4] = (int)(tile_d1 & 0xFFFFu);              // tile_dim1; tile_dim2=0 (2D)
    g1[5] = (int)(stride0_elems & 0xFFFFFFFFull);  // tensor_dim0_stride[31:0]
    g1[6] = (int)((stride0_elems >> 32) & 0xFFFFull); // [47:32]; dim1_stride=0
    g1[7] = 0;
    v4i gz4 = {0, 0, 0, 0};                        // groups 2/3 unused (2D tile)
    v8i gz8 = {0, 0, 0, 0, 0, 0, 0, 0};
    __builtin_amdgcn_tensor_load_to_lds(g0, g1, gz4, gz4, gz8, 0);
}
#endif

// ---------------------------------------------------------------------------
// Kernel A: lower-triangle tiles of S = X X^T + eps*I   (fp32 WMMA SYRK)
// grid = (17, 256 batches), block = 256 (8 waves); 17*8 = 136 = #lower tiles
// ---------------------------------------------------------------------------
__global__ __launch_bounds__(256) void gram_syrk_kernel(
    const float* __restrict__ X, float* __restrict__ S)
{
    const int b    = blockIdx.y;
    const int lane = threadIdx.x & 31;
    const int wave = threadIdx.x >> 5;
    int tile = blockIdx.x * 8 + wave;            // 0..135, lower-tri tile id

    // tile -> (ti, tj) with ti >= tj
    int ti = (int)((sqrtf(8.0f * (float)tile + 1.0f) - 1.0f) * 0.5f);
    while ((ti + 1) * (ti + 2) / 2 <= tile) ++ti;
    while (ti * (ti + 1) / 2 > tile) --ti;
    const int tj = tile - ti * (ti + 1) / 2;

    const int half = lane >> 4;                  // lane-half selects K msbs
    const int l15  = lane & 15;

    const float* xb   = X + (size_t)b * MDIM * KDIM;
    // fp32 A layout: lane l<16 holds (M=l, K=k..k+1); lanes 16-31 hold K=k+2..k+3
    const float* arow = xb + (size_t)(ti * 16 + l15) * KDIM + 2 * half;
    // B = X^T slice: B[k'][n] = X[tj*16+n][k+k'] -> identical float2 gather
    const float* brow = xb + (size_t)(tj * 16 + l15) * KDIM + 2 * half;

    v8f c = {};
    for (int k0 = 0; k0 < KDIM; k0 += 64) {
        __builtin_prefetch(arow + k0 + 64, 0, 1);
        __builtin_prefetch(brow + k0 + 64, 0, 1);
        #pragma unroll
        for (int k = 0; k < 64; k += 4) {
            v2f a  = *(const v2f*)(arow + k0 + k);
            v2f bb = *(const v2f*)(brow + k0 + k);
            c = __builtin_amdgcn_wmma_f32_16x16x4_f32(
                    /*neg_a=*/false, a, /*neg_b=*/false, bb,
                    /*c_mod=*/(short)0, c, /*reuse_a=*/false, /*reuse_b=*/false);
        }
    }

    // C/D layout: VGPR r -> row r (lanes 0-15) / row r+8 (lanes 16-31), col=lane&15
    float* sb = S + (size_t)b * MDIM * MDIM;
    #pragma unroll
    for (int r = 0; r < 8; ++r) {
        int row = ti * 16 + r + 8 * half;
        int col = tj * 16 + l15;
        float v = c[r];
        if (row == col) v += EPS_C;
        sb[(size_t)row * MDIM + col] = v;
    }
}

// ---------------------------------------------------------------------------
// Kernel B: in-place Cholesky of lower triangle, packed in LDS (129 KB)
// grid = 256 batches, block = 256 (one row per thread)
// ---------------------------------------------------------------------------
__global__ __launch_bounds__(256) void cholesky_kernel(float* __restrict__ S)
{
    __shared__ float Ls[MDIM * (MDIM + 1) / 2];  // packed: (i,j) at i(i+1)/2 + j
    const int b   = blockIdx.x;
    const int tid = threadIdx.x;
    float* sb = S + (size_t)b * MDIM * MDIM;

    // load lower triangle (coalesced along each row)
    for (int i = 0; i < MDIM; ++i) {
        const int base = i * (i + 1) / 2;
        for (int j = tid; j <= i; j += 256)
            Ls[base + j] = sb[(size_t)i * MDIM + j];
    }
    __syncthreads();

    const int i     = tid;
    const int ibase = i * (i + 1) / 2;
    int kbase = 0;                               // k(k+1)/2
    for (int k = 0; k < MDIM; ++k) {
        if (tid == 0) Ls[kbase + k] = sqrtf(Ls[kbase + k]);
        __syncthreads();
        const float inv = 1.0f / Ls[kbase + k];  // broadcast read
        if (i > k) Ls[ibase + k] *= inv;
        __syncthreads();
        if (i > k) {
            const float lik = Ls[ibase + k];
            int jb = kbase + (k + 1);            // (k+1)(k+2)/2
            for (int j = k + 1; j <= i; ++j) {
                Ls[ibase + j] -= lik * Ls[jb + k];  // Ls[jb+k]=L[j][k], broadcast
                jb += (j + 1);
            }
        }
        kbase += k + 1;
        __syncthreads();
    }

    // write L back over S (full row-major layout, lower triangle)
    for (int ii = 0; ii < MDIM; ++ii) {
        const int base = ii * (ii + 1) / 2;
        for (int j = tid; j <= ii; j += 256)
            sb[(size_t)ii * MDIM + j] = Ls[base + j];
    }
}

// ---------------------------------------------------------------------------
// Kernel C: left-looking blocked TRSM  L W = X  (16-row blocks, WMMA updates,
// TDM staging of the X block + Ltt tile into LDS)
// grid = 256 batches, block = 256 (8 waves)
// ---------------------------------------------------------------------------
__global__ __launch_bounds__(256) void trsm_kernel(
    const float* __restrict__ X, const float* __restrict__ L,
    float* __restrict__ W)
{
    __shared__ float rhs[16 * KDIM];   // 32 KB: X block -> accumulated RHS
    __shared__ float Ltt[16][16];
    __shared__ float invD[16];

    const int b    = blockIdx.x;
    const int tid  = threadIdx.x;
    const int lane = tid & 31;
    const int wave = tid >> 5;
    const int half = lane >> 4;
    const int l15  = lane & 15;

    const float* xb = X + (size_t)b * MDIM * KDIM;
    const float* lb = L + (size_t)b * MDIM * MDIM;
    float*       wb = W + (size_t)b * MDIM * KDIM;

    for (int t = 0; t < 16; ++t) {
        const int rowBase = t * 16;

#ifdef HAVE_TDM
        if (wave == 0) {
            // X[rowBase..rowBase+15][0..511] -> rhs  (row-major, contiguous)
            tdm_load_2d_f32((unsigned)(size_t)(void*)rhs,
                            xb + (size_t)rowBase * KDIM,
                            KDIM, 16, KDIM, 16, KDIM);
            // L[rowBase..+15][rowBase..+15] -> Ltt (upper junk never read)
            tdm_load_2d_f32((unsigned)(size_t)(void*)&Ltt[0][0],
                            lb + (size_t)rowBase * MDIM + rowBase,
                            16, 16, 16, 16, MDIM);
            __builtin_amdgcn_s_wait_tensorcnt(0);
        }
#else
        for (int idx = tid; idx < 16 * KDIM; idx += 256)
            rhs[idx] = xb[(size_t)rowBase * KDIM + idx];
        {
            const int r = tid >> 4, q = tid & 15;
            Ltt[r][q] = lb[(size_t)(rowBase + r) * MDIM + rowBase + q];
        }
#endif
        __syncthreads();
        if (tid < 16) invD[tid] = 1.0f / Ltt[tid][tid];

        // RHS = X[rows t] - L[rows t, 0:16t) * W[0:16t)   (fp32 WMMA, A negated)
        for (int nt = wave; nt < 32; nt += 8) {
            const int colBase = nt * 16;
            v8f c;
            #pragma unroll
            for (int r = 0; r < 8; ++r)
                c[r] = rhs[(r + 8 * half) * KDIM + colBase + l15];

            for (int kb = 0; kb < t; ++kb) {
                #pragma unroll
                for (int s = 0; s < 4; ++s) {
                    const int k0 = kb * 16 + s * 4 + 2 * half;
                    const float* ap = lb + (size_t)(rowBase + l15) * MDIM + k0;
                    v2f a;  a.x = -ap[0];  a.y = -ap[1];   // F32 WMMA: no NEG on A
                    const float* bp = wb + (size_t)k0 * KDIM + colBase + l15;
                    v2f bv; bv.x = bp[0];  bv.y = bp[KDIM];
                    c = __builtin_amdgcn_wmma_f32_16x16x4_f32(
                            false, a, false, bv, (short)0, c, false, false);
                }
            }
            #pragma unroll
            for (int r = 0; r < 8; ++r)
                rhs[(r + 8 * half) * KDIM + colBase + l15] = c[r];
        }
        __syncthreads();

        // solve 16x16 lower-triangular system for 512 columns (2 per thread)
        #pragma unroll
        for (int cc = 0; cc < 2; ++cc) {
            const int col = tid + cc * 256;
            float wv[16];
            #pragma unroll
            for (int r = 0; r < 16; ++r) {
                float v = rhs[r * KDIM + col];
                #pragma unroll
                for (int q = 0; q < r; ++q) v -= Ltt[r][q] * wv[q];
                wv[r] = v * invD[r];
            }
            #pragma unroll
            for (int r = 0; r < 16; ++r)
                wb[(size_t)(rowBase + r) * KDIM + col] = wv[r];
        }
        __syncthreads();   // W block visible to next t (CUMODE: one CU/workgroup)
    }
}

// ---------------------------------------------------------------------------
extern "C" void kernel_launch(void* const* d_in, const int* in_sizes, int n_in,
                              void* d_out, int out_size, void* d_ws, size_t ws_size,
                              hipStream_t stream)
{
    const float* X = (const float*)d_in[0];
    float* Wout = (float*)d_out;
    float* S = (float*)d_ws;   // 256*256*256*4 = 64 MiB scratch (L2-resident)

    hipLaunchKernelGGL(gram_syrk_kernel, dim3(17, NBATCH), dim3(256), 0, stream, X, S);
    hipLaunchKernelGGL(cholesky_kernel,  dim3(NBATCH),     dim3(256), 0, stream, S);
    hipLaunchKernelGGL(trsm_kernel,      dim3(NBATCH),     dim3(256), 0, stream, X, S, Wout);
}